// GaussianImage_Cholesky_HSI_83081847374188
// MI455X (gfx1250) — compile-verified
//
#include <hip/hip_runtime.h>
#include <math.h>

// Problem constants (from reference)
#define HH     256
#define WW     256
#define NN     4096
#define RANKK  4
#define CC     200
#define CPAD   208     // pad E columns to a multiple of 16 (13 WMMA N-tiles)
#define CHUNKB 128     // gaussians staged in LDS per iteration

typedef __attribute__((ext_vector_type(2))) float v2f;
typedef __attribute__((ext_vector_type(8))) float v8f;

// ---------------------------------------------------------------------------
// Phase A: per-gaussian preprocessing -> gp[n*16 + k] packed params
//   [cx, cy, ca, cb] [cc, f0, f1, f2] [f3, g0x, g0y, g1x] [g1y, w0, w1, 0]
// ---------------------------------------------------------------------------
__global__ void gi_prep_kernel(const float* __restrict__ xyz_raw,
                               const float* __restrict__ chol,
                               const float* __restrict__ opac,
                               const float* __restrict__ feat_raw,
                               const float* __restrict__ gfreq,
                               const float* __restrict__ gwt,
                               float* __restrict__ gp) {
    int n = blockIdx.x * blockDim.x + threadIdx.x;
    if (n >= NN) return;
    float tx = tanhf(xyz_raw[2 * n + 0]);
    float ty = tanhf(xyz_raw[2 * n + 1]);
    float cx = (tx + 1.0f) * 0.5f * (float)(WW - 1);
    float cy = (ty + 1.0f) * 0.5f * (float)(HH - 1);
    float l1 = chol[3 * n + 0] + 0.5f;
    float l2 = chol[3 * n + 1];
    float l3 = chol[3 * n + 2] + 0.5f;
    float a = l1 * l1;
    float b = l1 * l2;
    float c = l2 * l2 + l3 * l3;
    float det = fmaxf(a * c - b * b, 1e-12f);
    float rd = 1.0f / det;
    float ca = c * rd, cb = -b * rd, cc = a * rd;
    float op = opac[n];
    float f[RANKK];
#pragma unroll
    for (int r = 0; r < RANKK; ++r) {
        float x = feat_raw[RANKK * n + r];
        float sp = (x > 20.0f) ? x : log1pf(expf(x));   // softplus
        f[r] = sp * op;
    }
    float g0x = expf(gfreq[(2 * n + 0) * 2 + 0]);
    float g0y = expf(gfreq[(2 * n + 0) * 2 + 1]);
    float g1x = expf(gfreq[(2 * n + 1) * 2 + 0]);
    float g1y = expf(gfreq[(2 * n + 1) * 2 + 1]);
    float w0 = 1.0f / (1.0f + expf(-gwt[2 * n + 0]));   // sigmoid
    float w1 = 1.0f / (1.0f + expf(-gwt[2 * n + 1]));
    float4* o = (float4*)(gp + (size_t)n * 16);
    o[0] = make_float4(cx, cy, ca, cb);
    o[1] = make_float4(cc, f[0], f[1], f[2]);
    o[2] = make_float4(f[3], g0x, g0y, g1x);
    o[3] = make_float4(g1y, w0, w1, 0.0f);
}

// ---------------------------------------------------------------------------
// Phase C: E = max(E0 + 0.1*tanh(logit)*tanh(U@V), 1e-6), padded to 208 cols
// ---------------------------------------------------------------------------
__global__ void gi_lora_kernel(const float* __restrict__ U,
                               const float* __restrict__ V,
                               const float* __restrict__ logit,
                               const float* __restrict__ E0,
                               float* __restrict__ Epad) {
    int idx = blockIdx.x * blockDim.x + threadIdx.x;
    if (idx >= RANKK * CPAD) return;
    int r = idx / CPAD;
    int c = idx - r * CPAD;
    float val = 0.0f;
    if (c < CC) {
        float scale = 0.1f * tanhf(logit[0]);
        float d = tanhf(fmaf(U[r * 2 + 0], V[c], U[r * 2 + 1] * V[CC + c]));
        val = fmaxf(fmaf(scale, d, E0[r * CC + c]), 1e-6f);
    }
    Epad[r * CPAD + c] = val;
}

// ---------------------------------------------------------------------------
// Phase B: abundance[pix][r] = sum_n g(n,pix)*mod(n,pix)*feats[n][r]
// One pixel per lane; gaussian params staged through LDS in 8KB chunks.
// exp(p) = v_exp_f32(p*log2e); cos(2*pi*t) = v_cos_f32(t - rint(t))
// (v_cos_f32 argument is in revolutions -> exact period-1 range reduction).
// ---------------------------------------------------------------------------
__global__ __launch_bounds__(256) void gi_splat_kernel(
        const float* __restrict__ gp, float* __restrict__ abund) {
    __shared__ float4 sh[CHUNKB * 4];   // 8 KB
    const int tid = threadIdx.x;
    const int pix = blockIdx.x * 256 + tid;
    const float px = (float)(pix & (WW - 1));
    const float py = (float)(pix >> 8);
    float acc0 = 0.0f, acc1 = 0.0f, acc2 = 0.0f, acc3 = 0.0f;
    const float4* gp4 = (const float4*)gp;
    for (int base = 0; base < NN; base += CHUNKB) {
        __syncthreads();
#pragma unroll
        for (int i = 0; i < (CHUNKB * 4) / 256; ++i)
            sh[tid + i * 256] = gp4[(size_t)base * 4 + tid + i * 256];
        __syncthreads();
#pragma unroll 4
        for (int j = 0; j < CHUNKB; ++j) {
            float4 p0 = sh[j * 4 + 0];
            float4 p1 = sh[j * 4 + 1];
            float4 p2 = sh[j * 4 + 2];
            float4 p3 = sh[j * 4 + 3];
            float dx = px - p0.x;
            float dy = py - p0.y;
            // q = a*dx^2 + 2b*dx*dy + c*dy^2  (positive definite)
            float q = fmaf(dx, fmaf(2.0f * p0.w, dy, p0.z * dx), p1.x * dy * dy);
            float pw = fminf(-0.5f * q, 0.0f);
            float g = __builtin_amdgcn_exp2f(pw * 1.4426950408889634f);
            float t0 = fmaf(p2.y, dx, p2.z * dy);
            float t1 = fmaf(p2.w, dx, p3.x * dy);
            t0 -= rintf(t0);
            t1 -= rintf(t1);
            float c0 = __builtin_amdgcn_cosf(t0);
            float c1 = __builtin_amdgcn_cosf(t1);
            float mod = fmaf(p3.y, c0, fmaf(p3.z, c1, 1.0f));
            float wv = g * mod;
            acc0 = fmaf(wv, p1.y, acc0);
            acc1 = fmaf(wv, p1.z, acc1);
            acc2 = fmaf(wv, p1.w, acc2);
            acc3 = fmaf(wv, p2.x, acc3);
        }
    }
    ((float4*)abund)[pix] = make_float4(acc0, acc1, acc2, acc3);
}

// ---------------------------------------------------------------------------
// Phase D: out[hw][c] = abundance[hw][0:4] @ E[0:4][c] via v_wmma_f32_16x16x4_f32
// One wave per 16-row tile; loop 13 column tiles of Epad.
// A layout (ISA 7.12.2, 32-bit A 16x4): lane<16 -> K={0,1}, lane>=16 -> K={2,3}
// B layout mirrors A: lane = N (0..15), VGPR j + 2*(lane>=16) = K
// D layout: VGPR j, lane l -> (M = j + 8*(l>=16), N = l&15)
// ---------------------------------------------------------------------------
__global__ __launch_bounds__(256) void gi_project_kernel(
        const float* __restrict__ abund, const float* __restrict__ Epad,
        float* __restrict__ out) {
    const int lane = threadIdx.x & 31;
    const int wave = threadIdx.x >> 5;
    const int rowBase = (blockIdx.x * 8 + wave) * 16;
    const int half = lane >> 4;     // 0 or 1
    const int l15 = lane & 15;
    const int kb = half * 2;        // K base for this half-wave

    // A-matrix: abundance rows rowBase..rowBase+15, K = kb, kb+1 (8B load)
    float2 a2 = *(const float2*)(abund + (size_t)(rowBase + l15) * RANKK + kb);
    v2f A;
    A.x = a2.x;
    A.y = a2.y;

    for (int t = 0; t < CPAD / 16; ++t) {
        int col = t * 16 + l15;
        v2f B;
        B.x = Epad[(kb + 0) * CPAD + col];
        B.y = Epad[(kb + 1) * CPAD + col];
        v8f Cacc = {};
        v8f D = __builtin_amdgcn_wmma_f32_16x16x4_f32(
            /*neg_a=*/false, A, /*neg_b=*/false, B,
            /*c_mod=*/(short)0, Cacc, /*reuse_a=*/false, /*reuse_b=*/false);
        if (col < CC) {
#pragma unroll
            for (int j = 0; j < 8; ++j) {
                int m = j + half * 8;
                out[(size_t)(rowBase + m) * CC + col] = D[j];
            }
        }
    }
}

// ---------------------------------------------------------------------------
extern "C" void kernel_launch(void* const* d_in, const int* in_sizes, int n_in,
                              void* d_out, int out_size, void* d_ws, size_t ws_size,
                              hipStream_t stream) {
    const float* xyz_raw  = (const float*)d_in[0];
    const float* chol     = (const float*)d_in[1];
    const float* opac     = (const float*)d_in[2];
    const float* feat_raw = (const float*)d_in[3];
    const float* gfreq    = (const float*)d_in[4];
    const float* gwt      = (const float*)d_in[5];
    const float* lora_U   = (const float*)d_in[6];
    const float* lora_V   = (const float*)d_in[7];
    const float* logit    = (const float*)d_in[8];
    const float* E0       = (const float*)d_in[9];
    float* out = (float*)d_out;

    // Workspace layout (floats): gp[NN*16] | Epad[1024] | abund[HH*WW*4]
    float* gp    = (float*)d_ws;
    float* Epad  = gp + (size_t)NN * 16;
    float* abund = Epad + 1024;   // 16B-aligned (offset 266240 bytes)

    gi_prep_kernel<<<NN / 256, 256, 0, stream>>>(xyz_raw, chol, opac, feat_raw,
                                                 gfreq, gwt, gp);
    gi_lora_kernel<<<(RANKK * CPAD + 255) / 256, 256, 0, stream>>>(
        lora_U, lora_V, logit, E0, Epad);
    gi_splat_kernel<<<(HH * WW) / 256, 256, 0, stream>>>(gp, abund);
    gi_project_kernel<<<(HH * WW / 16) / 8, 256, 0, stream>>>(abund, Epad, out);
}